// RelGCN_82042465288592
// MI455X (gfx1250) — compile-verified
//
#include <hip/hip_runtime.h>
#include <hip/hip_bf16.h>

typedef __attribute__((ext_vector_type(2))) float v2f;
typedef __attribute__((ext_vector_type(8))) float v8f;

#define NNODES 50000
#define HID    128
#define NREL   8
#define NEDGE  600000
#define NBINS  (NNODES * NREL)          // 400000 (dst,rel) bins
#define SCANB  ((NBINS + 255) / 256)    // 1563 scan blocks

// ---------------------------------------------------------------- zero fill
__global__ void zero_f4(float4* p, long n4) {
  long i = (long)blockIdx.x * blockDim.x + threadIdx.x;
  long stride = (long)gridDim.x * blockDim.x;
  float4 z = {0.f, 0.f, 0.f, 0.f};
  for (; i < n4; i += stride) p[i] = z;
}

// --------------------------------------------- per-(dst,rel) in-edge counts
__global__ void count_kernel(const int* __restrict__ dst,
                             const int* __restrict__ rel,
                             int* __restrict__ cnt, int ne) {
  int e = blockIdx.x * blockDim.x + threadIdx.x;
  if (e >= ne) return;
  atomicAdd(&cnt[dst[e] * NREL + rel[e]], 1);
}

// ------------------------------------------------ exclusive scan, 3 kernels
__global__ void scan_blocksum(const int* __restrict__ cnt, int* __restrict__ bsum, int nb) {
  __shared__ int sh[256];
  int i = blockIdx.x * 256 + threadIdx.x;
  sh[threadIdx.x] = (i < nb) ? cnt[i] : 0;
  __syncthreads();
  for (int d = 128; d > 0; d >>= 1) {
    if (threadIdx.x < d) sh[threadIdx.x] += sh[threadIdx.x + d];
    __syncthreads();
  }
  if (threadIdx.x == 0) bsum[blockIdx.x] = sh[0];
}

__global__ void scan_bsum(int* __restrict__ bsum, int n) {
  __shared__ int sh[256];
  __shared__ int base;
  if (threadIdx.x == 0) base = 0;
  __syncthreads();
  for (int s = 0; s < n; s += 256) {
    int i = s + threadIdx.x;
    int v = (i < n) ? bsum[i] : 0;
    sh[threadIdx.x] = v;
    __syncthreads();
    for (int d = 1; d < 256; d <<= 1) {
      int t = (threadIdx.x >= d) ? sh[threadIdx.x - d] : 0;
      __syncthreads();
      sh[threadIdx.x] += t;
      __syncthreads();
    }
    int incl = sh[threadIdx.x];
    if (i < n) bsum[i] = base + incl - v;  // exclusive
    __syncthreads();
    if (threadIdx.x == 255) base += incl;  // incl of lane 255 == chunk total
    __syncthreads();
  }
}

__global__ void scan_final(const int* __restrict__ cnt, const int* __restrict__ bsum,
                           int* __restrict__ offs, int nb) {
  __shared__ int sh[256];
  int i = blockIdx.x * 256 + threadIdx.x;
  int v = (i < nb) ? cnt[i] : 0;
  sh[threadIdx.x] = v;
  __syncthreads();
  for (int d = 1; d < 256; d <<= 1) {
    int t = (threadIdx.x >= d) ? sh[threadIdx.x - d] : 0;
    __syncthreads();
    sh[threadIdx.x] += t;
    __syncthreads();
  }
  if (i < nb) offs[i] = bsum[blockIdx.x] + sh[threadIdx.x] - v;
}

// ----------------------------------- CSR fill: esrc[offs[bin]+slot] = src[e]
__global__ void fill_kernel(const int* __restrict__ src, const int* __restrict__ dst,
                            const int* __restrict__ rel, const int* __restrict__ offs,
                            int* __restrict__ cursor, int* __restrict__ esrc, int ne) {
  int e = blockIdx.x * blockDim.x + threadIdx.x;
  if (e >= ne) return;
  int bin = dst[e] * NREL + rel[e];
  int slot = atomicAdd(&cursor[bin], 1);
  esrc[offs[bin] + slot] = src[e];
}

// -------- gather: one wave per (dst,rel) bin; mean-normalized sum of x[src]
// Writes every bin (zeros included) -> pre needs no init, no float atomics.
__global__ void gather_kernel(const float* __restrict__ x, const int* __restrict__ esrc,
                              const int* __restrict__ offs, const int* __restrict__ cnt,
                              float* __restrict__ pre, int nbins) {
  int gid = blockIdx.x * blockDim.x + threadIdx.x;
  int w = gid >> 5;
  int lane = threadIdx.x & 31;
  if (w >= nbins) return;
  int start = offs[w];
  int c = cnt[w];
  float ax = 0.f, ay = 0.f, az = 0.f, aw = 0.f;
  for (int i = 0; i < c; ++i) {
    int s = esrc[start + i];
    float4 v = reinterpret_cast<const float4*>(x + (size_t)s * HID)[lane];
    ax += v.x; ay += v.y; az += v.z; aw += v.w;
  }
  float inv = (c > 0) ? (1.0f / (float)c) : 0.0f;  // 1/max(cnt,1)
  float4 r = {ax * inv, ay * inv, az * inv, aw * inv};
  reinterpret_cast<float4*>(pre + (size_t)w * HID)[lane] = r;
}

// ------------------------------------------------------------- fused GEMM
// out[n,:] = sum_r pre[n,r,:] @ W[r] + xin[n,:] @ root + bias  (pre is
// already mean-normalized). M: 16 rows/wave, 8 waves/block; N = 128 (8 WMMA
// column tiles); K = 9 blocks of 128 (8 relations + root), staged 64 rows at
// a time into LDS in WMMA-B fragment layout. B staging is software-pipelined:
// next sub-tile's global loads issue before the current tile's compute.
__global__ __launch_bounds__(256) void rgcn_gemm(
    const float* __restrict__ pre, const float* __restrict__ xin,
    const float* __restrict__ W, const float* __restrict__ root,
    const float* __restrict__ bias, float* __restrict__ outp, int relu) {
  __shared__ float lds[64 * HID];  // 32 KB swizzled B sub-tile
  const int tid = threadIdx.x;
  const int lane = tid & 31;
  const int wv = tid >> 5;  // 0..7
  const int rowbase = (blockIdx.x * 8 + wv) * 16;
  const bool active = rowbase < NNODES;
  const int rowA = (active ? rowbase : 0) + (lane & 15);

  v8f acc[8] = {};
  float4 g[8];  // staged B sub-tile (registers)

  auto loadTile = [&](int t) {  // t = blk*2 + sub, 0..17
    const float* wsrc = (t < 16) ? (W + (size_t)(t >> 1) * HID * HID) : root;
    const float4* g4 = reinterpret_cast<const float4*>(wsrc + (t & 1) * 64 * HID);
#pragma unroll
    for (int i = 0; i < 8; ++i) g[i] = g4[tid + i * 256];
  };

  loadTile(0);
  for (int t = 0; t < 18; ++t) {
    __syncthreads();  // previous compute done before LDS overwrite
    // store staged regs into fragment-swizzled LDS layout:
    // element (k,n): kq=k>>2, klo=k&3 -> lane=(n&15)+((klo&2)?16:0), v=klo&1
#pragma unroll
    for (int i = 0; i < 8; ++i) {
      int idx = tid + i * 256;  // 0..2047 float4s
      int k = idx >> 5;
      int n4 = (idx & 31) << 2;
      int kq = k >> 2, klo = k & 3;
      int hi = (klo & 2) ? 16 : 0;
      int v = klo & 1;
      float vv[4] = {g[i].x, g[i].y, g[i].z, g[i].w};
#pragma unroll
      for (int c = 0; c < 4; ++c) {
        int n = n4 + c;
        lds[((((kq << 3) + (n >> 4)) << 5) + ((n & 15) + hi)) * 2 + v] = vv[c];
      }
    }
    __syncthreads();  // staging visible

    if (t < 17) loadTile(t + 1);  // prefetch next B sub-tile under compute

    // A fragment: lanes 0-15 hold (K0,K1), lanes 16-31 hold (K2,K3)
    const int blk = t >> 1, sub = t & 1;
    const float* abase =
        ((blk < 8) ? (pre + (size_t)rowA * (NREL * HID) + blk * HID)
                   : (xin + (size_t)rowA * HID)) +
        sub * 64 + ((lane >> 4) << 1);
    const float2* bl = reinterpret_cast<const float2*>(lds);
#pragma unroll 4
    for (int kq = 0; kq < 16; ++kq) {
      float2 a2 = *reinterpret_cast<const float2*>(abase + kq * 4);
      v2f av = {a2.x, a2.y};
#pragma unroll
      for (int j = 0; j < 8; ++j) {
        float2 b2 = bl[((kq << 3) + j) * 32 + lane];
        v2f bv = {b2.x, b2.y};
        acc[j] = __builtin_amdgcn_wmma_f32_16x16x4_f32(
            false, av, false, bv, (short)0, acc[j], false, false);
      }
    }
  }

  if (active) {
    // C/D layout: VGPR i: lanes 0-15 -> M=i, lanes 16-31 -> M=8+i; N = lane&15
    int m0 = (lane >> 4) * 8;
    int nn = lane & 15;
#pragma unroll
    for (int j = 0; j < 8; ++j) {
      float bv = bias[j * 16 + nn];
#pragma unroll
      for (int i = 0; i < 8; ++i) {
        float v = acc[j][i] + bv;
        if (relu) v = fmaxf(v, 0.0f);
        outp[(size_t)(rowbase + m0 + i) * HID + j * 16 + nn] = v;
      }
    }
  }
}

// ---------------------------------------------------------------- launcher
extern "C" void kernel_launch(void* const* d_in, const int* in_sizes, int n_in,
                              void* d_out, int out_size, void* d_ws, size_t ws_size,
                              hipStream_t stream) {
  const int* edge_index = (const int*)d_in[0];
  const int* edge_type = (const int*)d_in[1];
  const float* node_emb = (const float*)d_in[2];
  const float* W1 = (const float*)d_in[3];
  const float* root1 = (const float*)d_in[4];
  const float* b1 = (const float*)d_in[5];
  const float* W2 = (const float*)d_in[6];
  const float* root2 = (const float*)d_in[7];
  const float* b2 = (const float*)d_in[8];
  const int* srcv = edge_index;          // edge_index[0]
  const int* dstv = edge_index + NEDGE;  // edge_index[1]

  char* ws = (char*)d_ws;
  size_t off = 0;
  float* pre = (float*)(ws + off); off += (size_t)NBINS * HID * 4;   // 204.8 MB
  float* h   = (float*)(ws + off); off += (size_t)NNODES * HID * 4;  // 25.6 MB
  int* cnt    = (int*)(ws + off); off += (size_t)NBINS * 4;          // 1.6 MB
  int* offs   = (int*)(ws + off); off += (size_t)NBINS * 4;          // 1.6 MB
  int* cursor = (int*)(ws + off); off += (size_t)NBINS * 4;          // 1.6 MB
  int* esrc   = (int*)(ws + off); off += (size_t)NEDGE * 4;          // 2.4 MB
  int* bsum   = (int*)(ws + off); off += 8192;
  float* out = (float*)d_out;

  const long cnt4 = (long)NBINS / 4;
  const int edgeBlocks = (NEDGE + 255) / 256;
  const int gatherBlocks = (NBINS * 32 + 255) / 256;  // one wave per bin
  const int gemmBlocks = (NNODES / 16 + 7) / 8;       // 391

  // ---- CSR over (dst,rel) bins: built once, shared by both layers
  zero_f4<<<512, 256, 0, stream>>>((float4*)cnt, cnt4);
  count_kernel<<<edgeBlocks, 256, 0, stream>>>(dstv, edge_type, cnt, NEDGE);
  scan_blocksum<<<SCANB, 256, 0, stream>>>(cnt, bsum, NBINS);
  scan_bsum<<<1, 256, 0, stream>>>(bsum, SCANB);
  scan_final<<<SCANB, 256, 0, stream>>>(cnt, bsum, offs, NBINS);
  zero_f4<<<512, 256, 0, stream>>>((float4*)cursor, cnt4);
  fill_kernel<<<edgeBlocks, 256, 0, stream>>>(srcv, dstv, edge_type, offs, cursor, esrc, NEDGE);

  // ---- layer 1
  gather_kernel<<<gatherBlocks, 256, 0, stream>>>(node_emb, esrc, offs, cnt, pre, NBINS);
  rgcn_gemm<<<gemmBlocks, 256, 0, stream>>>(pre, node_emb, W1, root1, b1, h, 1);

  // ---- layer 2
  gather_kernel<<<gatherBlocks, 256, 0, stream>>>(h, esrc, offs, cnt, pre, NBINS);
  rgcn_gemm<<<gemmBlocks, 256, 0, stream>>>(pre, h, W2, root2, b2, out, 0);
}